// TopologicalAutoencoderLoss_44727789421015
// MI455X (gfx1250) — compile-verified
//
#include <hip/hip_runtime.h>
#include <hip/hip_bf16.h>
#include <math.h>

// ---------------------------------------------------------------------------
// Topological autoencoder loss for MI455X (gfx1250, wave32, WMMA).
// Gram matrices via V_WMMA_F32_16X16X4_F32 (exact fp32 on the matrix pipe).
// Each wave computes a 16x64 strip: one A fragment reused against 4 B
// fragments -> 4 independent accumulators (hides WMMA RAW latency, raises
// FLOP/byte from ~4 to ~6.4 out of L2).
// ---------------------------------------------------------------------------

#define N_PTS   4096
#define D_DATA  1024
#define D_LAT   64
#define REG_LAMBDA 0.5f

typedef __attribute__((ext_vector_type(2))) float v2f;
typedef __attribute__((ext_vector_type(8))) float v8f;

// ---------------------------------------------------------------------------
// A fragment (16x4 fp32): lane l<16 holds row M=l, K = k0/k0+1 in 2 VGPRs;
//                         lane l>=16 holds row M=l-16, K = k0+2/k0+3.
// B fragment (4x16 fp32): B[k][n] = X[n][k] -> same load pattern as A but with
//                         the column-tile row base (gram symmetry).
// Block = 128 threads (4 waves); each wave owns 4 consecutive column tiles.
// ---------------------------------------------------------------------------
__global__ void __launch_bounds__(128)
gram_dist_wmma(const float* __restrict__ X, const float* __restrict__ sq,
               float* __restrict__ dist, int D, const float* __restrict__ norm_ptr)
{
    const int lane  = threadIdx.x & 31;
    const int wid   = threadIdx.x >> 5;
    const int half  = lane >> 4;       // 0: K pair (0,1)  1: K pair (2,3)
    const int l15   = lane & 15;

    const int tile_m  = blockIdx.y;                    // row tile
    const int tile_n0 = (blockIdx.x * 4 + wid) * 4;    // first of 4 column tiles

    const int m0 = tile_m * 16;
    const int n0 = tile_n0 * 16;

    const float* a_base  = X + (size_t)(m0 + l15) * D + 2 * half;
    const float* b_base0 = X + (size_t)(n0 +  0 + l15) * D + 2 * half;
    const float* b_base1 = X + (size_t)(n0 + 16 + l15) * D + 2 * half;
    const float* b_base2 = X + (size_t)(n0 + 32 + l15) * D + 2 * half;
    const float* b_base3 = X + (size_t)(n0 + 48 + l15) * D + 2 * half;

    v8f c0 = {}, c1 = {}, c2 = {}, c3 = {};

    for (int k0 = 0; k0 < D; k0 += 64) {
        // speculative prefetch of the next A-row chunk (global_prefetch_b8);
        // OOB speculative prefetches are silently dropped.
        __builtin_prefetch(a_base + k0 + 64, 0, 3);
#pragma unroll 4
        for (int kk = 0; kk < 64; kk += 4) {
            const int k = k0 + kk;
            v2f a  = *(const v2f*)(a_base  + k);
            v2f b0 = *(const v2f*)(b_base0 + k);
            v2f b1 = *(const v2f*)(b_base1 + k);
            v2f b2 = *(const v2f*)(b_base2 + k);
            v2f b3 = *(const v2f*)(b_base3 + k);
            // 8 args: (neg_a, A, neg_b, B, c_mod, C, reuse_a, reuse_b)
            c0 = __builtin_amdgcn_wmma_f32_16x16x4_f32(false, a, false, b0, (short)0, c0, false, false);
            c1 = __builtin_amdgcn_wmma_f32_16x16x4_f32(false, a, false, b1, (short)0, c1, false, false);
            c2 = __builtin_amdgcn_wmma_f32_16x16x4_f32(false, a, false, b2, (short)0, c2, false, false);
            c3 = __builtin_amdgcn_wmma_f32_16x16x4_f32(false, a, false, b3, (short)0, c3, false, false);
        }
    }

    const float scale = norm_ptr ? (1.0f / norm_ptr[0]) : 1.0f;
    const float sqrow[1] = {0.0f}; (void)sqrow;

    // C/D layout: VGPR r, lanes 0-15 -> M = r, lanes 16-31 -> M = r + 8; N = lane&15
#pragma unroll
    for (int j = 0; j < 4; ++j) {
        v8f c = (j == 0) ? c0 : (j == 1) ? c1 : (j == 2) ? c2 : c3;
        const int colbase = n0 + 16 * j + l15;
#pragma unroll
        for (int r = 0; r < 8; ++r) {
            const int row = m0 + r + 8 * half;
            float g  = c[r];
            float d2 = sq[row] + sq[colbase] - 2.0f * g;
            float d  = (d2 > 0.0f) ? sqrtf(d2) : 0.0f;
            dist[(size_t)row * N_PTS + colbase] = d * scale;
        }
    }
}

// ---------------------------------------------------------------------------
// Row squared norms: one block per row.
// ---------------------------------------------------------------------------
__global__ void __launch_bounds__(256)
row_sqnorm(const float* __restrict__ X, float* __restrict__ sq, int D)
{
    __shared__ float s[256];
    const int row = blockIdx.x;
    const float* p = X + (size_t)row * D;
    float acc = 0.0f;
    for (int i = threadIdx.x; i < D; i += 256) { float v = p[i]; acc += v * v; }
    s[threadIdx.x] = acc;
    __syncthreads();
    for (int off = 128; off > 0; off >>= 1) {
        if (threadIdx.x < off) s[threadIdx.x] += s[threadIdx.x + off];
        __syncthreads();
    }
    if (threadIdx.x == 0) sq[row] = s[0];
}

// ---------------------------------------------------------------------------
// Per-row MSE between y_true and y_pred (keras MSE, mean over last axis).
// ---------------------------------------------------------------------------
__global__ void __launch_bounds__(256)
ae_mse(const float* __restrict__ yt, const float* __restrict__ yp,
       float* __restrict__ ae)
{
    __shared__ float s[256];
    const int row = blockIdx.x;
    const float* a = yt + (size_t)row * D_DATA;
    const float* b = yp + (size_t)row * D_DATA;
    float acc = 0.0f;
    for (int i = threadIdx.x; i < D_DATA; i += 256) {
        float d = a[i] - b[i];
        acc += d * d;
    }
    s[threadIdx.x] = acc;
    __syncthreads();
    for (int off = 128; off > 0; off >>= 1) {
        if (threadIdx.x < off) s[threadIdx.x] += s[threadIdx.x + off];
        __syncthreads();
    }
    if (threadIdx.x == 0) ae[row] = s[0] * (1.0f / (float)D_DATA);
}

// ---------------------------------------------------------------------------
// Global max of a non-negative float array (float bits monotonic -> atomicMax
// on unsigned).
// ---------------------------------------------------------------------------
__global__ void init_scalars(unsigned* __restrict__ maxbits)
{
    if (threadIdx.x == 0 && blockIdx.x == 0) *maxbits = 0u;
}

__global__ void __launch_bounds__(256)
max_reduce(const float* __restrict__ a, size_t n, unsigned* __restrict__ maxbits)
{
    float m = 0.0f;
    const size_t stride = (size_t)gridDim.x * 256;
    for (size_t i = (size_t)blockIdx.x * 256 + threadIdx.x; i < n; i += stride)
        m = fmaxf(m, a[i]);
    // wave32 reduce
    for (int off = 16; off > 0; off >>= 1)
        m = fmaxf(m, __shfl_xor(m, off, 32));
    __shared__ float s[8];
    const int lane = threadIdx.x & 31, wid = threadIdx.x >> 5;
    if (lane == 0) s[wid] = m;
    __syncthreads();
    if (threadIdx.x == 0) {
        for (int w = 1; w < 8; ++w) m = fmaxf(m, s[w]);
        atomicMax(maxbits, __float_as_uint(m));
    }
}

__global__ void __launch_bounds__(256)
scale_by_invmax(float* __restrict__ a, size_t n4, const unsigned* __restrict__ maxbits)
{
    const float inv = 1.0f / __uint_as_float(*maxbits);
    const size_t i = (size_t)blockIdx.x * 256 + threadIdx.x;
    if (i < n4) {
        float4* p = (float4*)a;
        float4 v = p[i];
        v.x *= inv; v.y *= inv; v.z *= inv; v.w *= inv;
        p[i] = v;
    }
}

// ---------------------------------------------------------------------------
// Prim's MST: single persistent workgroup (1024 threads = 32 waves), state in
// LDS. Per step: parallel argmin (first-index tie-break, matches jnp.argmin),
// then relax with row dist[j].
// ---------------------------------------------------------------------------
__device__ inline void argmin32(float& v, int& i)
{
    for (int off = 16; off > 0; off >>= 1) {
        float ov = __shfl_down(v, off, 32);
        int   oi = __shfl_down(i, off, 32);
        if (ov < v || (ov == v && oi < i)) { v = ov; i = oi; }
    }
}

__global__ void __launch_bounds__(1024)
prim_mst(const float* __restrict__ dist, int* __restrict__ edges)
{
    __shared__ float    md[N_PTS];            // min distance to tree
    __shared__ int      ms[N_PTS];            // tree node achieving it
    __shared__ unsigned intree[N_PTS / 32];
    __shared__ float    sv[32];
    __shared__ int      si[32];
    __shared__ int      jsel;

    const int t    = threadIdx.x;
    const int lane = t & 31;
    const int wid  = t >> 5;
    const float INF = __int_as_float(0x7f800000);

    for (int i = t; i < N_PTS; i += 1024) { md[i] = dist[i]; ms[i] = 0; }
    for (int i = t; i < N_PTS / 32; i += 1024) intree[i] = 0u;
    __syncthreads();
    if (t == 0) intree[0] = 1u;   // node 0 seeds the tree
    __syncthreads();

    for (int step = 0; step < N_PTS - 1; ++step) {
        // ---- argmin over nodes not in tree ----
        float bv = INF; int bi = 0x7fffffff;
        for (int i = t; i < N_PTS; i += 1024) {
            bool in = (intree[i >> 5] >> (i & 31)) & 1u;
            float v = in ? INF : md[i];
            if (v < bv || (v == bv && i < bi)) { bv = v; bi = i; }
        }
        argmin32(bv, bi);
        if (lane == 0) { sv[wid] = bv; si[wid] = bi; }
        __syncthreads();
        if (wid == 0) {
            float v = sv[lane];
            int   i = si[lane];
            argmin32(v, i);
            if (lane == 0) jsel = i;
        }
        __syncthreads();
        const int j = jsel;
        if (t == 0) {
            edges[2 * step + 0] = ms[j];
            edges[2 * step + 1] = j;
            intree[j >> 5] |= (1u << (j & 31));
        }
        __syncthreads();
        // ---- relax with row dist[j] ----
        const float* dj = dist + (size_t)j * N_PTS;
        for (int i = t; i < N_PTS; i += 1024) {
            float v = dj[i];
            if (v < md[i]) { md[i] = v; ms[i] = j; }
        }
        __syncthreads();
    }
}

// ---------------------------------------------------------------------------
// mean((dd[e] - ld[e])^2) over MST edges; single block, direct store.
// ---------------------------------------------------------------------------
__global__ void __launch_bounds__(1024)
edge_loss(const float* __restrict__ dd, const float* __restrict__ ld,
          const int* __restrict__ edges, float* __restrict__ loss)
{
    __shared__ float s[1024];
    float acc = 0.0f;
    for (int e = threadIdx.x; e < N_PTS - 1; e += 1024) {
        const int i = edges[2 * e + 0];
        const int j = edges[2 * e + 1];
        const size_t k = (size_t)i * N_PTS + j;
        float d = dd[k] - ld[k];
        acc += d * d;
    }
    s[threadIdx.x] = acc;
    __syncthreads();
    for (int off = 512; off > 0; off >>= 1) {
        if (threadIdx.x < off) s[threadIdx.x] += s[threadIdx.x + off];
        __syncthreads();
    }
    if (threadIdx.x == 0) *loss = s[0] * (1.0f / (float)(N_PTS - 1));
}

// ---------------------------------------------------------------------------
// out[i] = ae[i] + REG_LAMBDA * (data_loss + latent_loss)
// ---------------------------------------------------------------------------
__global__ void __launch_bounds__(256)
finalize(const float* __restrict__ ae, const float* __restrict__ l0,
         const float* __restrict__ l1, float* __restrict__ out)
{
    const int i = blockIdx.x * 256 + threadIdx.x;
    if (i < N_PTS) out[i] = ae[i] + REG_LAMBDA * (l0[0] + l1[0]);
}

// ---------------------------------------------------------------------------
extern "C" void kernel_launch(void* const* d_in, const int* in_sizes, int n_in,
                              void* d_out, int out_size, void* d_ws, size_t ws_size,
                              hipStream_t stream)
{
    const float* y_true      = (const float*)d_in[0];   // [N, D_DATA]
    const float* latent      = (const float*)d_in[1];   // [N, D_LAT]
    const float* y_pred      = (const float*)d_in[2];   // [N, D_DATA]
    const float* latent_norm = (const float*)d_in[3];   // scalar

    float* out = (float*)d_out;                         // [N]

    // ---- workspace layout ----
    float*    dist_data = (float*)d_ws;                              // N*N
    float*    dist_lat  = dist_data + (size_t)N_PTS * N_PTS;         // N*N
    float*    ae        = dist_lat + (size_t)N_PTS * N_PTS;          // N
    float*    sqd       = ae + N_PTS;                                // N
    float*    sql       = sqd + N_PTS;                               // N
    unsigned* maxbits   = (unsigned*)(sql + N_PTS);                  // 1
    float*    loss0     = (float*)(maxbits + 1);                     // 1
    float*    loss1     = loss0 + 1;                                 // 1
    int*      edges_d   = (int*)(loss1 + 1);                         // 2*(N-1)
    int*      edges_l   = edges_d + 2 * (N_PTS - 1);                 // 2*(N-1)

    init_scalars<<<1, 32, 0, stream>>>(maxbits);

    row_sqnorm<<<N_PTS, 256, 0, stream>>>(y_true, sqd, D_DATA);
    row_sqnorm<<<N_PTS, 256, 0, stream>>>(latent, sql, D_LAT);
    ae_mse<<<N_PTS, 256, 0, stream>>>(y_true, y_pred, ae);

    // data gram/dist (unnormalized), then max-normalize
    {
        // each block: 4 waves x 4 column tiles = 16 column tiles
        dim3 grid((N_PTS / 16) / 16, N_PTS / 16);
        gram_dist_wmma<<<grid, 128, 0, stream>>>(y_true, sqd, dist_data, D_DATA, nullptr);
    }
    const size_t nn = (size_t)N_PTS * N_PTS;
    max_reduce<<<1024, 256, 0, stream>>>(dist_data, nn, maxbits);
    scale_by_invmax<<<(unsigned)((nn / 4 + 255) / 256), 256, 0, stream>>>(dist_data, nn / 4, maxbits);

    // latent gram/dist scaled by 1/latent_norm
    {
        dim3 grid((N_PTS / 16) / 16, N_PTS / 16);
        gram_dist_wmma<<<grid, 128, 0, stream>>>(latent, sql, dist_lat, D_LAT, latent_norm);
    }

    // MSTs (serial Prim per matrix, independent of each other)
    prim_mst<<<1, 1024, 0, stream>>>(dist_data, edges_d);
    prim_mst<<<1, 1024, 0, stream>>>(dist_lat, edges_l);

    edge_loss<<<1, 1024, 0, stream>>>(dist_data, dist_lat, edges_d, loss0);
    edge_loss<<<1, 1024, 0, stream>>>(dist_data, dist_lat, edges_l, loss1);

    finalize<<<(N_PTS + 255) / 256, 256, 0, stream>>>(ae, loss0, loss1, out);
}